// GaussianSplattingCompliance_49667001811229
// MI455X (gfx1250) — compile-verified
//
#include <hip/hip_runtime.h>
#include <hip/hip_bf16.h>
#include <math.h>

typedef __attribute__((ext_vector_type(2))) float v2f;
typedef __attribute__((ext_vector_type(8))) float v8f;

#define LOG2E 1.4426950408889634f

__device__ __forceinline__ float fast_sigmoid(float x) {
  // 1/(1+exp(-x)) = 1/(1+exp2(-x*log2e))
  return __builtin_amdgcn_rcpf(1.0f + __builtin_amdgcn_exp2f(-LOG2E * x));
}

// ---------------------------------------------------------------------------
// Kernel 1: min/max over coords (N,2) -> ws[0..3] = {minx, miny, maxx, maxy}
// ---------------------------------------------------------------------------
__global__ __launch_bounds__(256) void gsc_minmax(const float* __restrict__ coords,
                                                  int n, float* __restrict__ ws) {
  __shared__ float sminx[256], sminy[256], smaxx[256], smaxy[256];
  int tid = threadIdx.x;
  float mnx = 3.0e38f, mny = 3.0e38f, mxx = -3.0e38f, mxy = -3.0e38f;
  for (int i = tid; i < n; i += 256) {
    float x = coords[2 * i];
    float y = coords[2 * i + 1];
    mnx = fminf(mnx, x); mny = fminf(mny, y);
    mxx = fmaxf(mxx, x); mxy = fmaxf(mxy, y);
  }
  sminx[tid] = mnx; sminy[tid] = mny; smaxx[tid] = mxx; smaxy[tid] = mxy;
  __syncthreads();
  for (int s = 128; s > 0; s >>= 1) {
    if (tid < s) {
      sminx[tid] = fminf(sminx[tid], sminx[tid + s]);
      sminy[tid] = fminf(sminy[tid], sminy[tid + s]);
      smaxx[tid] = fmaxf(smaxx[tid], smaxx[tid + s]);
      smaxy[tid] = fmaxf(smaxy[tid], smaxy[tid + s]);
    }
    __syncthreads();
  }
  if (tid == 0) {
    ws[0] = sminx[0]; ws[1] = sminy[0]; ws[2] = smaxx[0]; ws[3] = smaxy[0];
  }
}

// ---------------------------------------------------------------------------
// Kernel 2: per-gaussian quadratic coefficients Q[g][8]
//   z^2 = q0*px^2 + q1*py^2 + q2*px*py + q3*px + q4*py + q5   (q6=q7=0 pad)
// Padded rows (g >= n_g) get q5 = 1e30 -> z huge -> kernel contribution = 0.
// ---------------------------------------------------------------------------
__global__ __launch_bounds__(256) void gsc_coeffs(const float* __restrict__ W_scale,
                                                  const float* __restrict__ W_shape,
                                                  const float* __restrict__ W_rot,
                                                  const float* __restrict__ W_off,
                                                  const float* __restrict__ bounds,
                                                  float* __restrict__ Q,
                                                  int n_g, int n_pad) {
  float cminx = bounds[0], cminy = bounds[1], cmaxx = bounds[2], cmaxy = bounds[3];
  float lo0 = cminx - cmaxx * 0.05f, hi0 = cmaxx + cmaxx * 0.05f;
  float lo1 = cminy - cmaxy * 0.05f, hi1 = cmaxy + cmaxy * 0.05f;

  for (int g = threadIdx.x; g < n_pad; g += 256) {
    float q0 = 0.f, q1 = 0.f, q2 = 0.f, q3 = 0.f, q4 = 0.f, q5 = 1.0e30f;
    if (g < n_g) {
      float bs  = 1.45f * fast_sigmoid(W_scale[g]) + 0.05f;   // base scale
      float si  = __builtin_amdgcn_rcpf(bs + 1e-8f);
      float si2 = si * si;
      float ratio = 3.0f * fast_sigmoid(W_shape[g]) + 0.5f;
      float icd0 = __builtin_amdgcn_rcpf(1.0f + 1e-6f);
      float icd1 = __builtin_amdgcn_rcpf(ratio * ratio + 1e-6f);
      float rot = -1.5707963267948966f + 3.141592653589793f * fast_sigmoid(W_rot[g]);
      float cr = __cosf(rot);
      float sr = __sinf(rot);
      float ox = lo0 + (hi0 - lo0) * fast_sigmoid(W_off[2 * g]);
      float oy = lo1 + (hi1 - lo1) * fast_sigmoid(W_off[2 * g + 1]);
      float a = cr * cr * icd0 + sr * sr * icd1;
      float b = sr * sr * icd0 + cr * cr * icd1;
      float c = 2.0f * cr * sr * (icd1 - icd0);
      q0 = si2 * a;
      q1 = si2 * b;
      q2 = si2 * c;
      q3 = si2 * (-2.0f * a * ox - c * oy);
      q4 = si2 * (-2.0f * b * oy - c * ox);
      q5 = si2 * (a * ox * ox + b * oy * oy + c * ox * oy);
    }
    float* q = Q + g * 8;
    q[0] = q0; q[1] = q1; q[2] = q2; q[3] = q3;
    q[4] = q4; q[5] = q5; q[6] = 0.f; q[7] = 0.f;
  }
}

// ---------------------------------------------------------------------------
// Kernel 3: main splat pass. Each wave handles 16 points x all gaussians.
//   A (16x4 f32, per gaussian tile) = coefficients, B (4x16 f32) = point feats.
//   D tile (16 gaussians x 16 points) via two chained v_wmma_f32_16x16x4_f32.
//   Q table staged to LDS with async direct-to-LDS copies (ASYNCcnt path).
//   Tile loop software-pipelined: A operands for t+1 fetched while the
//   transcendental chain for tile t executes (hides ds latency).
// ---------------------------------------------------------------------------
__global__ __launch_bounds__(256) void gsc_main(const float* __restrict__ coords,
                                                const float* __restrict__ Q,
                                                float* __restrict__ out,
                                                int n_points, int ntiles) {
  __shared__ float Qs[256 * 8];  // up to 16 gaussian tiles (8 KB)
  int ng16 = ntiles * 16;

  // Stage coefficient table global -> LDS with async b128 copies (no VGPR hop).
  // GVS mode: LDS-address VGPR, 32-bit byte-offset VGPR, SGPR base pair.
  for (int i = threadIdx.x; i < ng16 * 2; i += 256) {
    unsigned int ldsaddr = (unsigned int)(uintptr_t)&Qs[i * 4];
    unsigned int goff = (unsigned int)(i * 16);
    asm volatile("global_load_async_to_lds_b128 %0, %1, %2"
                 :: "v"(ldsaddr), "v"(goff), "s"(Q)
                 : "memory");
  }
  asm volatile("s_wait_asynccnt 0x0" ::: "memory");
  __syncthreads();

  int lane = threadIdx.x & 31;
  int wave = threadIdx.x >> 5;
  int pn   = lane & 15;   // point (N) index within tile
  int half = lane >> 4;   // 0: K rows {0,1}/{4,5}; 1: K rows {2,3}/{6,7}
  int pointBase = (blockIdx.x * 8 + wave) * 16;
  int pidx = pointBase + pn;
  int pclamp = pidx < n_points ? pidx : n_points - 1;  // keep EXEC all-ones

  float px = coords[2 * pclamp];
  float py = coords[2 * pclamp + 1];
  // feature vector f = {px^2, py^2, px*py, px, py, 1, 0, 0}
  float f0 = px * px, f1 = py * py, f2 = px * py;
  // B-matrix (4x16): VGPR0 holds K=0 (lanes 0-15) / K=2 (lanes 16-31);
  //                  VGPR1 holds K=1 / K=3. Second WMMA covers K=4..7.
  v2f bmat_lo = { half ? f2 : f0, half ? px : f1 };
  v2f bmat_hi = { half ? 0.f : py, half ? 0.f : 1.0f };

  int qsel = half * 2;
  // Prologue: fetch tile 0's A operands.
  const float* q0p = &Qs[pn * 8 + qsel];
  v2f alo = { q0p[0], q0p[1] };
  v2f ahi = { q0p[4], q0p[5] };

  float partial = 0.f;
  for (int t = 0; t < ntiles; ++t) {
    // Prefetch next tile's A operands (uniform branch; EXEC stays all-ones).
    v2f alo_n = alo, ahi_n = ahi;
    if (t + 1 < ntiles) {
      const float* qn = &Qs[((t + 1) * 16 + pn) * 8 + qsel];
      alo_n = v2f{ qn[0], qn[1] };
      ahi_n = v2f{ qn[4], qn[5] };
    }

    v8f acc = {};
    acc = __builtin_amdgcn_wmma_f32_16x16x4_f32(
        false, alo, false, bmat_lo, (short)0, acc, false, false);
    acc = __builtin_amdgcn_wmma_f32_16x16x4_f32(
        false, ahi, false, bmat_hi, (short)0, acc, false, false);

    float tsum = 0.f;
#pragma unroll
    for (int e = 0; e < 8; ++e) {
      float z2 = fmaxf(acc[e], 0.0f);
      float z  = __builtin_amdgcn_sqrtf(z2);
      // sigmoid(-10(z-1)) = 1/(1+exp2(10*log2e*(z-1)))
      float ex = __builtin_amdgcn_exp2f(10.0f * LOG2E * (z - 1.0f));
      tsum += __builtin_amdgcn_rcpf(1.0f + ex);
    }
    partial += tsum;

    alo = alo_n;
    ahi = ahi_n;
  }

  // Combine the two gaussian halves (M=0..7 in lanes 0-15, M=8..15 in 16-31).
  float total = partial + __shfl_xor(partial, 16, 32);
  float ks = total + 1e-8f;
  float H = (1.0f - 1e-9f) * fast_sigmoid(-10.0f * (ks - 0.5f)) + 1e-9f;
  if (half == 0 && pidx < n_points) out[pidx] = H;
}

// ---------------------------------------------------------------------------
extern "C" void kernel_launch(void* const* d_in, const int* in_sizes, int n_in,
                              void* d_out, int out_size, void* d_ws, size_t ws_size,
                              hipStream_t stream) {
  const float* coords  = (const float*)d_in[0];
  const float* W_scale = (const float*)d_in[1];
  const float* W_shape = (const float*)d_in[2];
  const float* W_rot   = (const float*)d_in[3];
  const float* W_off   = (const float*)d_in[4];
  float* out = (float*)d_out;

  int n_points = in_sizes[0] / 2;
  int n_g = in_sizes[1];
  int ntiles = (n_g + 15) / 16;
  if (ntiles > 16) ntiles = 16;   // LDS table sized for 256 gaussians
  int n_pad = ntiles * 16;

  float* wsf = (float*)d_ws;
  float* Q = wsf + 8;  // 32B-aligned coefficient table

  gsc_minmax<<<1, 256, 0, stream>>>(coords, n_points, wsf);
  gsc_coeffs<<<1, 256, 0, stream>>>(W_scale, W_shape, W_rot, W_off, wsf, Q,
                                    n_g, n_pad);
  int blocks = (n_points + 127) / 128;  // 8 waves x 16 points per block
  gsc_main<<<blocks, 256, 0, stream>>>(coords, Q, out, n_points, ntiles);
}